// GCNProp_23819888623645
// MI455X (gfx1250) — compile-verified
//
#include <hip/hip_runtime.h>
#include <stdint.h>

// GCN propagation: out = D^-1/2 (A_noself + I) D^-1/2 · X
// N=100000 nodes, E=1600000 edges, D=64 features.
// L2-resident scatter/gather workload -> optimize the data-movement path:
//   - TDM (tensor_load_to_lds) double-buffered async staging of edge chunks
//   - 16 lanes/edge float4 gathers (256B per row, fully coalesced)
//   - global_prefetch_b8 lookahead on gather rows
//   - global_atomic_add_f32 scatter

#define CHUNK 2048
#define AGG_BLOCK 256

typedef unsigned int uint32x4 __attribute__((ext_vector_type(4)));
typedef int          int32x8  __attribute__((ext_vector_type(8)));
typedef int          int32x4  __attribute__((ext_vector_type(4)));

__global__ void gcn_init_deg(unsigned* __restrict__ deg, int N) {
    int i = blockIdx.x * blockDim.x + threadIdx.x;
    if (i < N) deg[i] = 1u;  // self-loop contributes weight 1
}

__global__ void gcn_degree(const int* __restrict__ ei, unsigned* __restrict__ deg, int E) {
    int e = blockIdx.x * blockDim.x + threadIdx.x;
    if (e < E) {
        int r = ei[e];
        int c = ei[E + e];
        if (r != c) atomicAdd(&deg[r], 1u);  // self-loop edges have weight 0
    }
}

__global__ void gcn_dinv(const unsigned* __restrict__ deg, float* __restrict__ dinv, int N) {
    int i = blockIdx.x * blockDim.x + threadIdx.x;
    if (i < N) {
        float d = (float)deg[i];          // deg >= 1 always (self loop)
        dinv[i] = rsqrtf(d);
    }
}

// out[i,:] = dinv[i]^2 * x[i,:]  (self-loop term; also initializes d_out)
__global__ void gcn_self_init(const float4* __restrict__ x4, const float* __restrict__ dinv,
                              float4* __restrict__ out4, int n16) {
    int i = blockIdx.x * blockDim.x + threadIdx.x;  // over N*16 float4s
    if (i < n16) {
        int node = i >> 4;
        float s = dinv[node];
        s = s * s;
        float4 v = x4[i];
        float4 o;
        o.x = s * v.x; o.y = s * v.y; o.z = s * v.z; o.w = s * v.w;
        out4[i] = o;
    }
}

// LDS byte offset of a __shared__ object: flat LDS addresses map to the LDS
// aperture with the in-LDS offset in the low 32 bits (addr[31:0]).
__device__ __forceinline__ unsigned lds_offset_of(const void* p) {
    return (unsigned)(uintptr_t)p;
}

// Issue a TDM load of a CHUNK x 2 int32 tile (rows line + cols line, stride E)
// from the edge-index tensor into LDS at byte offset lds_off.
__device__ __forceinline__ void tdm_load_edges(const int* ei, int E, int chunk_start,
                                               unsigned lds_off) {
    uint64_t ga   = (uint64_t)(uintptr_t)(ei + chunk_start);  // tile start (rows line)
    unsigned dim0 = (unsigned)(E - chunk_start);              // OOB bound relative to tile

    uint32x4 g0;
    g0[0] = 1u;                                               // count=1 (valid), user mode
    g0[1] = lds_off;                                          // lds_addr (bytes)
    g0[2] = (unsigned)ga;                                     // global_addr[31:0]
    g0[3] = (unsigned)((ga >> 32) & 0x1FFFFFFull)             // global_addr[56:32]
          | (2u << 30);                                       // type=2 ("image")

    int32x8 g1;
    g1[0] = (int)(2u << 16);                                  // data_size=2 -> 4 bytes
    g1[1] = (int)((dim0 & 0xFFFFu) << 16);                    // tensor_dim0[15:0]
    g1[2] = (int)((dim0 >> 16) | (2u << 16));                 // tensor_dim0[31:16] | tensor_dim1=2
    g1[3] = (int)(((unsigned)CHUNK) << 16);                   // tile_dim0 = CHUNK
    g1[4] = 2;                                                // tile_dim1=2, tile_dim2=0
    g1[5] = E;                                                // tensor_dim0_stride = E dwords
    g1[6] = 0;                                                // stride hi / dim1_stride (unused, 2D)
    g1[7] = 0;

    int32x4 z4 = {0, 0, 0, 0};                                // groups 2/3 unused (<=2D)
    int32x8 z8 = {0, 0, 0, 0, 0, 0, 0, 0};                    // extra group (clang-23 arity)
    __builtin_amdgcn_tensor_load_to_lds(g0, g1, z4, z4, z8, 0);
}

__global__ void __launch_bounds__(AGG_BLOCK)
gcn_aggregate(const float* __restrict__ x, const int* __restrict__ ei,
              const float* __restrict__ dinv, float* __restrict__ out,
              int E, int nChunks) {
    __shared__ int sbuf[2][2 * CHUNK];   // [buffer][rows(CHUNK) ++ cols(CHUNK)]
    const float4* x4 = (const float4*)x;

    const int  sub      = threadIdx.x & 15;   // which float4 of the 64-wide row
    const int  grp      = threadIdx.x >> 4;   // edge slot within pass (16 edges/pass)
    const bool leadWave = (threadIdx.x < 32); // wave 0 drives the TDM pipeline

    int c0 = blockIdx.x;
    if (leadWave && c0 < nChunks)
        tdm_load_edges(ei, E, c0 * CHUNK, lds_offset_of(&sbuf[0][0]));

    int parity = 0;
    for (int ch = c0; ch < nChunks; ch += gridDim.x) {
        if (leadWave) __builtin_amdgcn_s_wait_tensorcnt(0);   // current buffer landed
        __syncthreads();
        // TDM wrote LDS behind the compiler's back: force re-load of sbuf.
        asm volatile("" ::: "memory");

        int nxt = ch + gridDim.x;                             // prefetch next chunk via TDM
        if (leadWave && nxt < nChunks)
            tdm_load_edges(ei, E, nxt * CHUNK, lds_offset_of(&sbuf[parity ^ 1][0]));

        const int* rows = sbuf[parity];
        const int* cols = sbuf[parity] + CHUNK;
        const int  base = ch * CHUNK;

        for (int i = grp; i < CHUNK; i += 16) {
            // software prefetch the gather row we will touch 4 passes from now
            int ip = i + 64;
            if (ip < CHUNK && base + ip < E) {
                int cp = cols[ip];
                __builtin_prefetch(&x4[(size_t)cp * 16 + sub], 0, 3);
            }

            int e = base + i;
            int r = rows[i];
            int c = cols[i];
            // OOB TDM reads give (0,0) -> filtered by r!=c; e<E guards garbage tails
            if (e < E && r != c) {
                float  nrm = dinv[r] * dinv[c];
                float4 v   = x4[(size_t)c * 16 + sub];        // global_load_b128, coalesced
                float* o   = out + ((size_t)r * 64 + (sub << 2));
                atomicAdd(o + 0, nrm * v.x);
                atomicAdd(o + 1, nrm * v.y);
                atomicAdd(o + 2, nrm * v.z);
                atomicAdd(o + 3, nrm * v.w);
            }
        }
        __syncthreads();   // all done reading this buffer before it is refilled
        parity ^= 1;
    }
}

extern "C" void kernel_launch(void* const* d_in, const int* in_sizes, int n_in,
                              void* d_out, int out_size, void* d_ws, size_t ws_size,
                              hipStream_t stream) {
    const float* x  = (const float*)d_in[0];
    const int*   ei = (const int*)d_in[1];     // edge_index [2, E], row-major
    const int N = in_sizes[0] / 64;
    const int E = in_sizes[1] / 2;
    float* out = (float*)d_out;

    unsigned* deg  = (unsigned*)d_ws;                               // N u32
    float*    dinv = (float*)((char*)d_ws + (size_t)N * sizeof(unsigned)); // N f32

    gcn_init_deg<<<(N + 255) / 256, 256, 0, stream>>>(deg, N);
    gcn_degree<<<(E + 255) / 256, 256, 0, stream>>>(ei, deg, E);
    gcn_dinv<<<(N + 255) / 256, 256, 0, stream>>>(deg, dinv, N);

    int n16 = N * 16;
    gcn_self_init<<<(n16 + 255) / 256, 256, 0, stream>>>(
        (const float4*)x, dinv, (float4*)out, n16);

    int nChunks = (E + CHUNK - 1) / CHUNK;
    int grid    = (nChunks + 1) / 2;   // 2 chunks/block -> exercises the double buffer
    gcn_aggregate<<<grid, AGG_BLOCK, 0, stream>>>(x, ei, dinv, out, E, nChunks);
}